// LKA_scalable_12240656794001
// MI455X (gfx1250) — compile-verified
//
#include <hip/hip_runtime.h>
#include <math.h>
#include <stdint.h>

// ---------------- CDNA5 WMMA fp32 16x16x4 helpers ----------------
typedef __attribute__((ext_vector_type(2))) float v2f;
typedef __attribute__((ext_vector_type(8))) float v8f;

__device__ __forceinline__ v8f wmma4(v2f a, v2f b, v8f c) {
  // D = A(16x4) x B(4x16) + C(16x16), fp32, wave32
  return __builtin_amdgcn_wmma_f32_16x16x4_f32(false, a, false, b, (short)0, c, false, false);
}

// ---------------- async global->LDS (CDNA5 ASYNCcnt path) ----------------
#if defined(__has_builtin)
#if __has_builtin(__builtin_amdgcn_global_load_async_to_lds_b32)
#define HAVE_ASYNC_LDS 1
#endif
#endif
#ifndef HAVE_ASYNC_LDS
#define HAVE_ASYNC_LDS 0
#endif

__device__ __forceinline__ void async_g2l_b32(const float* g, float* lds) {
#if HAVE_ASYNC_LDS
  __builtin_amdgcn_global_load_async_to_lds_b32(
      (__attribute__((address_space(1))) int*)(uintptr_t)g,
      (__attribute__((address_space(3))) int*)(uint32_t)(uintptr_t)lds, 0, 0);
#else
  *lds = *g;
#endif
}

__device__ __forceinline__ void wait_async_lds() {
#if HAVE_ASYNC_LDS
#if __has_builtin(__builtin_amdgcn_s_wait_asynccnt)
  __builtin_amdgcn_s_wait_asynccnt(0);
#else
  asm volatile("s_wait_asynccnt 0" ::: "memory");
#endif
#endif
}

// A fragment: M x K = 16 x 4, row-major source with leading dim lda.
// lanes 0-15: VGPR0=K0, VGPR1=K1 ; lanes 16-31: VGPR0=K2, VGPR1=K3 (ISA 7.12.2)
__device__ __forceinline__ v2f loadA(const float* A, int lda, int kb, int lane) {
  int m = lane & 15;
  int hk = (lane >> 4) << 1;
  const float* p = A + (size_t)m * lda + kb + hk;
  v2f r; r.x = p[0]; r.y = p[1];
  return r;
}

// B fragment: K x N = 4 x 16, row-major source with leading dim ldb (caller pre-offsets to n0)
__device__ __forceinline__ v2f loadB(const float* Bm, int ldb, int kb, int lane) {
  int n = lane & 15;
  int hk = (lane >> 4) << 1;
  v2f r;
  r.x = Bm[(size_t)(kb + hk) * ldb + n];
  r.y = Bm[(size_t)(kb + hk + 1) * ldb + n];
  return r;
}

// guarded B fragment (zero-fill columns >= nmax) -- only used on edge tiles
__device__ __forceinline__ v2f loadB_g(const float* Bm, int ldb, int kb, int n0, int nmax, int lane) {
  int n = n0 + (lane & 15);
  int hk = (lane >> 4) << 1;
  v2f r;
  if (n >= nmax) { r.x = 0.f; r.y = 0.f; return r; }
  r.x = Bm[(size_t)(kb + hk) * ldb + n];
  r.y = Bm[(size_t)(kb + hk + 1) * ldb + n];
  return r;
}

// C/D store: VGPR i -> row (i or i+8), col = lane&15 (ISA 7.12.2)
__device__ __forceinline__ void storeC_g(float* Cp, int ldc, int nvalid, v8f acc, int lane) {
  int n = lane & 15;
  if (n >= nvalid) return;
  int mo = (lane >> 4) * 8;
#pragma unroll
  for (int i = 0; i < 8; ++i) Cp[(size_t)(mo + i) * ldc + n] = acc[i];
}

// ---------------- problem constants ----------------
#define BB 4
#define CC 64
#define HH 256
#define WW 256
#define WF 129      // rfft width
#define WFP 132     // padded (multiple of 4) for irow K loop
#define NPIX (HH * WF)       // 33024
#define NPIXF (HH * WW)      // 65536
#define HPAD 258
#define WPAD 131

// ---------------- DFT / twiddle tables ----------------
__global__ void k_tables(float* rowc, float* rows_, float* colc, float* cols_,
                         float* irc, float* irs) {
  int idx = blockIdx.x * blockDim.x + threadIdx.x;
  const float TWO_PI = 6.283185307179586f;
  if (idx < 256 * WF) {  // forward row DFT (real -> half spectrum): e^{-i 2pi wk/256}
    int w = idx / WF, k = idx % WF;
    float th = TWO_PI * (float)((w * k) % 256) / 256.f;
    rowc[idx] = cosf(th);
    rows_[idx] = -sinf(th);
  }
  if (idx < 256 * 256) {  // symmetric col table: cos/-sin(2pi a b /256)
    int a = idx / 256, b = idx % 256;
    float th = TWO_PI * (float)((a * b) % 256) / 256.f;
    colc[idx] = cosf(th);
    cols_[idx] = -sinf(th);
  }
  if (idx < WFP * 256) {  // inverse row (hermitian fold + 1/256), zero-padded K rows
    int k = idx / 256, w = idx % 256;
    if (k < WF) {
      float wk = (k == 0 || k == 128) ? 1.f : 2.f;
      float th = TWO_PI * (float)((k * w) % 256) / 256.f;
      irc[idx] = wk * cosf(th) / 256.f;
      irs[idx] = -wk * sinf(th) / 256.f;
    } else {
      irc[idx] = 0.f;
      irs[idx] = 0.f;
    }
  }
}

// ---------------- rfft along rows: T[h,k] = sum_w x[h,w] * (c + i s) ----------------
__global__ void k_rfft_rows(const float* __restrict__ x, const float* __restrict__ rc,
                            const float* __restrict__ rs, float* tr, float* ti) {
  int lane = threadIdx.x;
  int img = blockIdx.y;  // b*C + c
  int tm = blockIdx.x / 9, tn = blockIdx.x % 9;
  int m0 = tm * 16, n0 = tn * 16;
  const float* A = x + (size_t)img * NPIXF + (size_t)m0 * WW;
  v8f ar = {}, ai = {};
  if (n0 + 16 <= WF) {  // full tile: branch-free WMMA pipeline
#pragma unroll 4
    for (int kb = 0; kb < 256; kb += 4) {
      v2f a = loadA(A, WW, kb, lane);
      v2f bc = loadB(rc + n0, WF, kb, lane);
      v2f bs = loadB(rs + n0, WF, kb, lane);
      ar = wmma4(a, bc, ar);
      ai = wmma4(a, bs, ai);
    }
  } else {  // edge tile: guarded B loads
#pragma unroll 4
    for (int kb = 0; kb < 256; kb += 4) {
      v2f a = loadA(A, WW, kb, lane);
      v2f bc = loadB_g(rc, WF, kb, n0, WF, lane);
      v2f bs = loadB_g(rs, WF, kb, n0, WF, lane);
      ar = wmma4(a, bc, ar);
      ai = wmma4(a, bs, ai);
    }
  }
  storeC_g(tr + (size_t)img * NPIX + (size_t)m0 * WF + n0, WF, WF - n0, ar, lane);
  storeC_g(ti + (size_t)img * NPIX + (size_t)m0 * WF + n0, WF, WF - n0, ai, lane);
}

// ---------------- complex DFT along columns (forward or inverse) ----------------
// tables store c = cos(th), s = -sin(th).
// forward: Or = c*Br - s*Bi ; Oi = c*Bi + s*Br   (e^{-i th})
// inverse: Or = c*Br + s*Bi ; Oi = c*Bi - s*Br   (e^{+i th}), scaled by 1/256
template <bool INV, bool EDGE>
__device__ __forceinline__ void dft_col_loop(const float* tcA, const float* tsA,
                                             const float* Br, const float* Bi, int n0, int lane,
                                             v8f& accP, v8f& accM1, v8f& accM2) {
#pragma unroll 4
  for (int kb = 0; kb < 256; kb += 4) {
    v2f ac = loadA(tcA, 256, kb, lane);
    v2f as = loadA(tsA, 256, kb, lane);
    v2f vr, vi;
    if (EDGE) {
      vr = loadB_g(Br, WF, kb, n0, WF, lane);
      vi = loadB_g(Bi, WF, kb, n0, WF, lane);
    } else {
      vr = loadB(Br + n0, WF, kb, lane);
      vi = loadB(Bi + n0, WF, kb, lane);
    }
    if (INV) {
      accP = wmma4(ac, vr, accP);
      accP = wmma4(as, vi, accP);
      accM1 = wmma4(ac, vi, accM1);
      accM2 = wmma4(as, vr, accM2);
    } else {
      accP = wmma4(ac, vi, accP);
      accP = wmma4(as, vr, accP);
      accM1 = wmma4(ac, vr, accM1);
      accM2 = wmma4(as, vi, accM2);
    }
  }
}

template <bool INV>
__global__ void k_dft_cols(const float* __restrict__ tc, const float* __restrict__ ts,
                           const float* __restrict__ br, const float* __restrict__ bi,
                           float* outr, float* outi, int out_ld, float scale) {
  int lane = threadIdx.x;
  int img = blockIdx.y;
  int tm = blockIdx.x / 9, tn = blockIdx.x % 9;
  int m0 = tm * 16, n0 = tn * 16;
  const float* Br = br + (size_t)img * NPIX;
  const float* Bi = bi + (size_t)img * NPIX;
  const float* tcA = tc + (size_t)m0 * 256;
  const float* tsA = ts + (size_t)m0 * 256;
  v8f accP = {}, accM1 = {}, accM2 = {};
  if (n0 + 16 <= WF)
    dft_col_loop<INV, false>(tcA, tsA, Br, Bi, n0, lane, accP, accM1, accM2);
  else
    dft_col_loop<INV, true>(tcA, tsA, Br, Bi, n0, lane, accP, accM1, accM2);
  v8f oR, oI;
#pragma unroll
  for (int i = 0; i < 8; ++i) {
    if (INV) { oR[i] = accP[i] * scale; oI[i] = (accM1[i] - accM2[i]) * scale; }
    else     { oR[i] = accM1[i] - accM2[i]; oI[i] = accP[i]; }
  }
  int nmax = INV ? WFP : WF;
  storeC_g(outr + (size_t)img * 256 * out_ld + (size_t)m0 * out_ld + n0, out_ld, nmax - n0, oR, lane);
  storeC_g(outi + (size_t)img * 256 * out_ld + (size_t)m0 * out_ld + n0, out_ld, nmax - n0, oI, lane);
}

// ---------------- amp / phase ----------------
__global__ void k_amppha(const float* yr, const float* yi, float* amp, float* pha, int n) {
  int i = blockIdx.x * blockDim.x + threadIdx.x;
  if (i >= n) return;
  float r = yr[i], im = yi[i];
  amp[i] = sqrtf(r * r + im * im);
  pha[i] = atan2f(im, r);
}

// ---------------- 3x3 offset conv: C=64 -> 18, pad 1 (direct VALU) ----------------
__global__ void k_offconv(const float* __restrict__ x, const float* __restrict__ wp,
                          const float* __restrict__ bp, float* off) {
  int idx = blockIdx.x * blockDim.x + threadIdx.x;
  const int total = BB * 18 * NPIX;
  if (idx >= total) return;
  int w = idx % WF;
  int h = (idx / WF) % HH;
  int oc = (idx / NPIX) % 18;
  int b = idx / (NPIX * 18);
  float acc = bp[oc];
  const float* xb = x + (size_t)b * CC * NPIX;
  const float* wo = wp + (size_t)oc * CC * 9;
  for (int c = 0; c < CC; ++c) {
    const float* xc = xb + (size_t)c * NPIX;
    if (c + 1 < CC) __builtin_prefetch(xc + NPIX + h * WF + w, 0, 1);  // global_prefetch_b8
    const float* w9 = wo + c * 9;
#pragma unroll
    for (int ki = 0; ki < 3; ++ki) {
      int hh = h + ki - 1;
      if ((unsigned)hh >= (unsigned)HH) continue;
#pragma unroll
      for (int kj = 0; kj < 3; ++kj) {
        int ww = w + kj - 1;
        if ((unsigned)ww >= (unsigned)WF) continue;
        acc += w9[ki * 3 + kj] * xc[hh * WF + ww];
      }
    }
  }
  off[idx] = acc;
}

// ---------------- fused deformable conv (sample -> LDS -> WMMA GEMM) ----------------
// out[b,oc,h,w] = sum_{di,dj} wc[oc,c,di,dj] * xo_big[3h-1+di, 3w-1+dj]
// tap (di,dj) reads sample index (i,j)=((di+2)%3,(dj+2)%3) at pixel shifted by -1 when di/dj==0
__global__ __launch_bounds__(256) void k_deform(const float* __restrict__ xin,
                                                const float* __restrict__ off,
                                                const float* __restrict__ wc, float* out) {
  __shared__ float Sm[64][32];
  __shared__ float Wt[64][64];
  __shared__ float g0[32], g1[32], g2[32], g3[32];
  __shared__ int ix0[32], iy0[32], ix1[32], iy1[32], sval[32];
  int t = threadIdx.x;
  int lane = t & 31, wv = t >> 5;
  int b = blockIdx.y;
  int pbase = blockIdx.x * 32;
  const float* xb = xin + (size_t)b * CC * NPIX;
  const float* ob = off + (size_t)b * 18 * NPIX;
  int mt = wv & 3, nt = wv >> 2;
  v8f acc = {};
  for (int di = 0; di < 3; ++di) {
    for (int dj = 0; dj < 3; ++dj) {
      int i = (di + 2) % 3, j = (dj + 2) % 3, k = i * 3 + j;
      __syncthreads();  // previous tap's WMMA reads done
      if (t < 32) {
        int pid = pbase + t;
        int oh = pid / WF, ow = pid % WF;
        int hh = oh + (di == 0 ? -1 : 0);
        int ww = ow + (dj == 0 ? -1 : 0);
        int valid = (hh >= 0) && (ww >= 0);
        float glt = 0, grb = 0, glb = 0, grt = 0;
        int qx0 = 0, qy0 = 0, qx1 = 0, qy1 = 0;
        if (valid) {
          float dx = ob[(size_t)k * NPIX + hh * WF + ww];
          float dy = ob[(size_t)(k + 9) * NPIX + hh * WF + ww];
          float px = (float)(hh + 1) + (float)(i - 1) + dx;
          float py = (float)(ww + 1) + (float)(j - 1) + dy;
          float fx = floorf(px), fy = floorf(py);
          float qx0f = fminf(fmaxf(fx, 0.f), (float)(HPAD - 1));
          float qy0f = fminf(fmaxf(fy, 0.f), (float)(WPAD - 1));
          float qx1f = fminf(fmaxf(fx + 1.f, 0.f), (float)(HPAD - 1));
          float qy1f = fminf(fmaxf(fy + 1.f, 0.f), (float)(WPAD - 1));
          float cpx = fminf(fmaxf(px, 0.f), (float)(HPAD - 1));
          float cpy = fminf(fmaxf(py, 0.f), (float)(WPAD - 1));
          glt = (1.f + (qx0f - cpx)) * (1.f + (qy0f - cpy));
          grb = (1.f - (qx1f - cpx)) * (1.f - (qy1f - cpy));
          glb = (1.f + (qx0f - cpx)) * (1.f - (qy1f - cpy));
          grt = (1.f - (qx1f - cpx)) * (1.f + (qy0f - cpy));
          qx0 = (int)qx0f; qy0 = (int)qy0f; qx1 = (int)qx1f; qy1 = (int)qy1f;
        }
        g0[t] = glt; g1[t] = grb; g2[t] = glb; g3[t] = grt;
        ix0[t] = qx0; iy0[t] = qy0; ix1[t] = qx1; iy1[t] = qy1; sval[t] = valid;
      }
      // stage this tap's 64x64 weight block asynchronously: Wt[oc][c] = wc[(oc*64+c)*9 + di*3+dj]
#pragma unroll
      for (int q = 0; q < 16; ++q) {
        int idx = t * 16 + q;
        async_g2l_b32(wc + (size_t)idx * 9 + (di * 3 + dj), &Wt[idx >> 6][idx & 63]);
      }
      wait_async_lds();
      __syncthreads();  // coords + weights visible
      // bilinear gather into Sm[c][p]
      {
        int p = t & 31;
        int cb = t >> 5;
        if (sval[p]) {
          float glt = g0[p], grb = g1[p], glb = g2[p], grt = g3[p];
          int x0 = ix0[p], y0 = iy0[p], x1 = ix1[p], y1 = iy1[p];
#pragma unroll
          for (int q = 0; q < 8; ++q) {
            int c = cb * 8 + q;
            const float* xc = xb + (size_t)c * NPIX;
            float v00 = (x0 >= 1 && x0 <= HH && y0 >= 1 && y0 <= WF) ? xc[(x0 - 1) * WF + (y0 - 1)] : 0.f;
            float v11 = (x1 >= 1 && x1 <= HH && y1 >= 1 && y1 <= WF) ? xc[(x1 - 1) * WF + (y1 - 1)] : 0.f;
            float v01 = (x0 >= 1 && x0 <= HH && y1 >= 1 && y1 <= WF) ? xc[(x0 - 1) * WF + (y1 - 1)] : 0.f;
            float v10 = (x1 >= 1 && x1 <= HH && y0 >= 1 && y0 <= WF) ? xc[(x1 - 1) * WF + (y0 - 1)] : 0.f;
            Sm[c][p] = glt * v00 + grb * v11 + glb * v01 + grt * v10;
          }
        } else {
#pragma unroll
          for (int q = 0; q < 8; ++q) Sm[cb * 8 + q][p] = 0.f;
        }
      }
      __syncthreads();  // samples visible
      // GEMM: acc(16oc x 16px) += Wt[mt] x Sm[:, nt], K = 64
#pragma unroll
      for (int kb = 0; kb < 64; kb += 4) {
        v2f a = loadA(&Wt[mt * 16][0], 64, kb, lane);
        v2f bf = loadB(&Sm[0][nt * 16], 32, kb, lane);
        acc = wmma4(a, bf, acc);
      }
    }
  }
  float* Cp = out + (size_t)b * CC * NPIX + (size_t)(mt * 16) * NPIX + pbase + nt * 16;
  int n = lane & 15, mo = (lane >> 4) * 8;
#pragma unroll
  for (int ii = 0; ii < 8; ++ii) Cp[(size_t)(mo + ii) * NPIX + n] = acc[ii];
}

// ---------------- 1x1 conv + bias via WMMA (async LDS staging) ----------------
__global__ __launch_bounds__(256) void k_conv1x1(const float* __restrict__ xin,
                                                 const float* __restrict__ w1,
                                                 const float* __restrict__ b1, float* out) {
  __shared__ float Sm[64][32];
  int t = threadIdx.x, lane = t & 31, wv = t >> 5;
  int b = blockIdx.y;
  int pbase = blockIdx.x * 32;
  const float* xb = xin + (size_t)b * CC * NPIX;
  {
    int p = t & 31, cb = t >> 5;
#pragma unroll
    for (int q = 0; q < 8; ++q) {
      int c = cb * 8 + q;
      async_g2l_b32(xb + (size_t)c * NPIX + pbase + p, &Sm[c][p]);
    }
    wait_async_lds();
  }
  __syncthreads();
  int mt = wv & 3, nt = wv >> 2;
  v8f acc = {};
#pragma unroll
  for (int kb = 0; kb < 64; kb += 4) {
    v2f a = loadA(w1 + (size_t)(mt * 16) * 64, 64, kb, lane);
    v2f bf = loadB(&Sm[0][nt * 16], 32, kb, lane);
    acc = wmma4(a, bf, acc);
  }
  float* Cp = out + (size_t)b * CC * NPIX + (size_t)(mt * 16) * NPIX + pbase + nt * 16;
  int n = lane & 15, mo = (lane >> 4) * 8;
#pragma unroll
  for (int ii = 0; ii < 8; ++ii)
    Cp[(size_t)(mo + ii) * NPIX + n] = acc[ii] + b1[mt * 16 + mo + ii];
}

// ---------------- combine (elementwise, trig on VALU) ----------------
__global__ void k_combine(const float* a3, const float* p3, const float* a1, const float* p1,
                          float* cr, float* ci, int n) {
  int i = blockIdx.x * blockDim.x + threadIdx.x;
  if (i >= n) return;
  float A3 = a3[i], P3 = p3[i], A1 = a1[i], P1 = p1[i];
  cr[i] = A1 * cosf(P3) + A3 * cosf(P1) + 3e-8f;  // real_13 + real_31 + eps
  ci[i] = A3 * sinf(P1) + A1 * sinf(P3) + 2e-8f;  // imag_31 + imag_13
}

// ---------------- inverse row DFT + abs: xr[h,w] = |sum_k Zr*irc + Zi*irs| ----------------
__global__ void k_irow_abs(const float* __restrict__ zr, const float* __restrict__ zi,
                           const float* __restrict__ irc, const float* __restrict__ irs,
                           float* xr) {
  int lane = threadIdx.x;
  int img = blockIdx.y;
  int tm = blockIdx.x >> 4, tn = blockIdx.x & 15;
  int m0 = tm * 16, n0 = tn * 16;
  const float* Ar = zr + (size_t)img * 256 * WFP + (size_t)m0 * WFP;
  const float* Ai = zi + (size_t)img * 256 * WFP + (size_t)m0 * WFP;
  v8f acc = {};
#pragma unroll 4
  for (int kb = 0; kb < WFP; kb += 4) {
    v2f ar = loadA(Ar, WFP, kb, lane);
    v2f ai = loadA(Ai, WFP, kb, lane);
    v2f bc = loadB(irc + n0, WW, kb, lane);
    v2f bs = loadB(irs + n0, WW, kb, lane);
    acc = wmma4(ar, bc, acc);
    acc = wmma4(ai, bs, acc);
  }
  float* Cp = xr + (size_t)img * NPIXF + (size_t)m0 * WW + n0;
  int n = lane & 15, mo = (lane >> 4) * 8;
#pragma unroll
  for (int ii = 0; ii < 8; ++ii) Cp[(size_t)(mo + ii) * WW + n] = fabsf(acc[ii]);
}

// ---------------- final 5x5 conv + bias via WMMA (async LDS staging) ----------------
__global__ __launch_bounds__(256) void k_conv5(const float* __restrict__ xin,
                                               const float* __restrict__ w0,
                                               const float* __restrict__ b0, float* out) {
  __shared__ float Sm[64][32];
  __shared__ float Wt[64][64];
  int t = threadIdx.x, lane = t & 31, wv = t >> 5;
  int b = blockIdx.y;
  int pbase = blockIdx.x * 32;
  const float* xb = xin + (size_t)b * CC * NPIXF;
  int mt = wv & 3, nt = wv >> 2;
  v8f acc = {};
  for (int tap = 0; tap < 25; ++tap) {
    int di = tap / 5 - 2, dj = tap % 5 - 2;
    __syncthreads();  // previous tap's WMMA reads done
#pragma unroll
    for (int q = 0; q < 16; ++q) {
      int idx = t * 16 + q;
      async_g2l_b32(w0 + (size_t)idx * 25 + tap, &Wt[idx >> 6][idx & 63]);
    }
    {
      int p = t & 31, cb = t >> 5;
      int pid = pbase + p;
      int oh = pid >> 8, ow = pid & 255;
      int hh = oh + di, ww = ow + dj;
      bool inb = (unsigned)hh < (unsigned)HH && (unsigned)ww < (unsigned)WW;
      if (inb) {
#pragma unroll
        for (int q = 0; q < 8; ++q) {
          int c = cb * 8 + q;
          async_g2l_b32(xb + (size_t)c * NPIXF + hh * WW + ww, &Sm[c][p]);
        }
      } else {
#pragma unroll
        for (int q = 0; q < 8; ++q) Sm[cb * 8 + q][p] = 0.f;
      }
    }
    wait_async_lds();
    __syncthreads();  // weights + samples visible
#pragma unroll
    for (int kb = 0; kb < 64; kb += 4) {
      v2f a = loadA(&Wt[mt * 16][0], 64, kb, lane);
      v2f bf = loadB(&Sm[0][nt * 16], 32, kb, lane);
      acc = wmma4(a, bf, acc);
    }
  }
  float* Cp = out + (size_t)b * CC * NPIXF + (size_t)(mt * 16) * NPIXF + pbase + nt * 16;
  int n = lane & 15, mo = (lane >> 4) * 8;
#pragma unroll
  for (int ii = 0; ii < 8; ++ii)
    Cp[(size_t)(mo + ii) * NPIXF + n] = acc[ii] + b0[mt * 16 + mo + ii];
}

// ---------------- host launcher ----------------
extern "C" void kernel_launch(void* const* d_in, const int* in_sizes, int n_in,
                              void* d_out, int out_size, void* d_ws, size_t ws_size,
                              hipStream_t stream) {
  const float* x    = (const float*)d_in[0];
  const float* wp_a = (const float*)d_in[1];
  const float* bp_a = (const float*)d_in[2];
  const float* wc_a = (const float*)d_in[3];
  const float* w1_a = (const float*)d_in[4];
  const float* b1_a = (const float*)d_in[5];
  const float* wp_p = (const float*)d_in[6];
  const float* bp_p = (const float*)d_in[7];
  const float* wc_p = (const float*)d_in[8];
  const float* w1_p = (const float*)d_in[9];
  const float* b1_p = (const float*)d_in[10];
  const float* w0   = (const float*)d_in[11];
  const float* b0   = (const float*)d_in[12];
  float* outp = (float*)d_out;

  // workspace layout (floats)
  const size_t P = (size_t)BB * CC * NPIX;       // 8,454,144
  const size_t OFFSZ = (size_t)BB * 18 * NPIX;   // 2,377,728
  const size_t IT = (size_t)BB * CC * 256 * WFP; // 8,650,752
  float* w = (float*)d_ws;
  float* t_rowc = w;
  float* t_rows = t_rowc + 256 * WF;
  float* t_colc = t_rows + 256 * WF;
  float* t_cols = t_colc + 256 * 256;
  float* t_irc  = t_cols + 256 * 256;
  float* t_irs  = t_irc + WFP * 256;
  float* amp  = t_irs + WFP * 256;   // later reused as cr
  float* pha  = amp + P;             // later reused as ci
  float* amp3 = pha + P;             // first used as row-DFT real temp
  float* pha3 = amp3 + P;            // first used as row-DFT imag temp
  float* amp1 = pha3 + P;            // first used as col-DFT real (Yr)
  float* pha1 = amp1 + P;            // first used as col-DFT imag (Yi)
  float* off_a = pha1 + P;
  float* off_p = off_a + OFFSZ;
  float* itr = off_p + OFFSZ;
  float* iti = itr + IT;
  float* xr  = iti + IT;

  const int nimg = BB * CC;  // 256 images

  // 1) twiddle tables
  k_tables<<<256, 256, 0, stream>>>(t_rowc, t_rows, t_colc, t_cols, t_irc, t_irs);

  // 2) rfft rows -> (amp3, pha3) used as temps
  k_rfft_rows<<<dim3(16 * 9, nimg), 32, 0, stream>>>(x, t_rowc, t_rows, amp3, pha3);

  // 3) forward col DFT -> (amp1, pha1) = (Yr, Yi)
  k_dft_cols<false><<<dim3(16 * 9, nimg), 32, 0, stream>>>(t_colc, t_cols, amp3, pha3,
                                                           amp1, pha1, WF, 1.0f);

  // 4) amplitude / phase
  {
    int n = (int)P;
    k_amppha<<<(n + 255) / 256, 256, 0, stream>>>(amp1, pha1, amp, pha, n);
  }

  // 5) offset convs
  {
    int tot = (int)OFFSZ;
    int g = (tot + 255) / 256;
    k_offconv<<<g, 256, 0, stream>>>(amp, wp_a, bp_a, off_a);
    k_offconv<<<g, 256, 0, stream>>>(pha, wp_p, bp_p, off_p);
  }

  // 6) deformable convs -> amp3, pha3 (temps now free)
  k_deform<<<dim3(NPIX / 32, BB), 256, 0, stream>>>(amp, off_a, wc_a, amp3);
  k_deform<<<dim3(NPIX / 32, BB), 256, 0, stream>>>(pha, off_p, wc_p, pha3);

  // 7) 1x1 convs -> amp1, pha1 (Yr/Yi now free)
  k_conv1x1<<<dim3(NPIX / 32, BB), 256, 0, stream>>>(amp, w1_a, b1_a, amp1);
  k_conv1x1<<<dim3(NPIX / 32, BB), 256, 0, stream>>>(pha, w1_p, b1_p, pha1);

  // 8) combine -> (cr, ci) in amp/pha slots (amp/pha no longer needed)
  {
    int n = (int)P;
    k_combine<<<(n + 255) / 256, 256, 0, stream>>>(amp3, pha3, amp1, pha1, amp, pha, n);
  }

  // 9) inverse col DFT (scale 1/256) -> itr, iti (row stride WFP, padded cols zeroed)
  k_dft_cols<true><<<dim3(16 * 9, nimg), 32, 0, stream>>>(t_colc, t_cols, amp, pha,
                                                          itr, iti, WFP, 1.0f / 256.0f);

  // 10) inverse row DFT + abs -> xr
  k_irow_abs<<<dim3(256, nimg), 32, 0, stream>>>(itr, iti, t_irc, t_irs, xr);

  // 11) final 5x5 conv + bias -> d_out
  k_conv5<<<dim3(NPIXF / 32, BB), 256, 0, stream>>>(xr, w0, b0, outp);

  (void)in_sizes; (void)n_in; (void)out_size; (void)ws_size;
}